// LIFNode_30769145709280
// MI455X (gfx1250) — compile-verified
//
#include <hip/hip_runtime.h>

// LIF forward scan, T=4 fixed by the reference.
// x_seq: [T, N] fp32 (N = B*197*768), out: [T, N] fp32 of {0,1} spikes.
// Memory-bound: ~620 MB streamed @ 23.3 TB/s -> ~27 us floor. No matmul
// structure -> no WMMA; optimize the data path instead (async LDS staging,
// b128 NT loads/stores).

typedef float v4f __attribute__((ext_vector_type(4)));
typedef int   v4i __attribute__((ext_vector_type(4)));

#if defined(__AMDGCN__)
typedef __attribute__((address_space(1))) v4i* gbl_v4i_ptr;  // global
typedef __attribute__((address_space(3))) v4i* lds_v4i_ptr;  // LDS
#endif

#define LIF_T 4
#define BLOCK 256

#if defined(__AMDGCN__) && __has_builtin(__builtin_amdgcn_global_load_async_to_lds_b128)
#define LIF_USE_ASYNC_LDS 1
#else
#define LIF_USE_ASYNC_LDS 0
#endif

__global__ __launch_bounds__(BLOCK) void lif_scan_kernel(
    const float* __restrict__ x, float* __restrict__ out, long long nPerT) {
  const int tid = (int)threadIdx.x;
  const long long chunk = (long long)blockIdx.x * BLOCK + tid;  // float4 index
  const long long base = chunk * 4;                             // element index
  const bool full = (base + 4 <= nPerT);

#if LIF_USE_ASYNC_LDS
  // Stage this block's 4 timestep tiles into LDS via the gfx1250 async
  // global->LDS engine (ASYNCcnt-tracked, bypasses VGPRs). Each lane stages
  // 16B per timestep and later reads back only its own slots -> no barrier.
  __shared__ v4f tile[LIF_T][BLOCK];
  if (full) {
#pragma unroll
    for (int t = 0; t < LIF_T; ++t) {
      const float* gsrc = x + (size_t)t * (size_t)nPerT + (size_t)base;
      __builtin_amdgcn_global_load_async_to_lds_b128(
          (gbl_v4i_ptr)gsrc, (lds_v4i_ptr)&tile[t][tid],
          /*imm offset*/ 0, /*cpol*/ 0);
    }
  }
#if __has_builtin(__builtin_amdgcn_s_wait_asynccnt)
  __builtin_amdgcn_s_wait_asynccnt(0);
#else
  asm volatile("s_wait_asynccnt 0" ::: "memory");
#endif
  asm volatile("" ::: "memory");  // keep LDS reads below the wait
#endif

  if (full) {
    v4f v = {0.0f, 0.0f, 0.0f, 0.0f};
#pragma unroll
    for (int t = 0; t < LIF_T; ++t) {
#if LIF_USE_ASYNC_LDS
      v4f xv = tile[t][tid];
#else
      const v4f* gsrc =
          (const v4f*)(x + (size_t)t * (size_t)nPerT + (size_t)base);
      __builtin_prefetch(gsrc, 0, 0);  // -> global_prefetch_b8 on gfx1250
      v4f xv = __builtin_nontemporal_load(gsrc);
#endif
      v4f sp;
#pragma unroll
      for (int i = 0; i < 4; ++i) {
        float vc = 0.5f * (v[i] + xv[i]);     // charge: v + (x - v)/TAU, TAU=2
        bool s = (vc >= 1.0f);                // fire: heaviside(vc - 1)
        sp[i] = s ? 1.0f : 0.0f;
        v[i] = s ? 0.0f : vc;                 // hard reset to V_RESET=0
      }
      // Streaming store: NT hint keeps the 620MB stream out of L2.
      __builtin_nontemporal_store(
          sp, (v4f*)(out + (size_t)t * (size_t)nPerT + (size_t)base));
    }
  } else if (base < nPerT) {
    // Generic tail (not hit for the reference shape: N % 1024 == 0).
    for (long long e = base; e < nPerT; ++e) {
      float v = 0.0f;
      for (int t = 0; t < LIF_T; ++t) {
        float vc = 0.5f * (v + x[(size_t)t * (size_t)nPerT + (size_t)e]);
        bool s = (vc >= 1.0f);
        out[(size_t)t * (size_t)nPerT + (size_t)e] = s ? 1.0f : 0.0f;
        v = s ? 0.0f : vc;
      }
    }
  }
}

extern "C" void kernel_launch(void* const* d_in, const int* in_sizes, int n_in,
                              void* d_out, int out_size, void* d_ws,
                              size_t ws_size, hipStream_t stream) {
  (void)n_in; (void)out_size; (void)d_ws; (void)ws_size;
  const float* x = (const float*)d_in[0];
  float* out = (float*)d_out;
  const long long total = (long long)in_sizes[0];   // T * N
  const long long nPerT = total / LIF_T;            // N per timestep
  const long long chunks = (nPerT + 3) / 4;         // float4 columns
  const long long grid = (chunks + BLOCK - 1) / BLOCK;
  lif_scan_kernel<<<dim3((unsigned int)grid), dim3(BLOCK), 0, stream>>>(
      x, out, nPerT);
}